// EdgeConvBlock_60859686584363
// MI455X (gfx1250) — compile-verified
//
#include <hip/hip_runtime.h>

typedef float v2f __attribute__((ext_vector_type(2)));
typedef float v8f __attribute__((ext_vector_type(8)));

#define N_NODES 50000
#define F_IN    64
#define F_OUT   64
#define N_EDGES 800000

// ---------------------------------------------------------------------------
// ws layout (floats):
//   Wc  : [128][64]        at offset 0            (combined weight, 32 KB)
//   y   : [N_NODES][128]   at offset 8192         (y1 = y[:, :64], y2 = y[:,64:])
//   cnt : [N_NODES]        after y
// ---------------------------------------------------------------------------

// Wc[n][k] = (n<64) ? W[n][k] - W[n][64+k]  :  W[n-64][64+k]
__global__ void prep_weights(const float* __restrict__ W, float* __restrict__ Wc) {
    int idx = blockIdx.x * blockDim.x + threadIdx.x;
    if (idx >= 128 * 64) return;
    int n = idx >> 6, k = idx & 63;
    if (n < 64)
        Wc[idx] = W[n * 128 + k] - W[n * 128 + 64 + k];
    else
        Wc[idx] = W[(n - 64) * 128 + 64 + k];
}

// zero the output accumulator and edge counts (must run every call)
__global__ void zero_bufs(float* __restrict__ out, float* __restrict__ cnt) {
    int idx = blockIdx.x * blockDim.x + threadIdx.x;
    if (idx < N_NODES * F_OUT) out[idx] = 0.0f;
    if (idx < N_NODES)         cnt[idx] = 0.0f;
}

// y[50000,128] = x[50000,64] @ Wc^T   via V_WMMA_F32_16X16X4_F32
// block = 256 threads = 8 waves; each block does a 16-row node tile,
// each wave one 16-wide N tile; K = 64 -> 16 WMMA steps of K=4.
__global__ __launch_bounds__(256) void gemm_xWc(const float* __restrict__ x,
                                                const float* __restrict__ Wc,
                                                float* __restrict__ y) {
    const int lane    = threadIdx.x & 31;
    const int wave    = threadIdx.x >> 5;        // 0..7 -> N tile
    const int rowBase = blockIdx.x * 16;         // 3125 blocks, no tail
    const int nBase   = wave * 16;
    const int l15     = lane & 15;
    const int hi      = lane >> 4;               // 0: K0/K1, 1: K2/K3
    const int kp      = hi * 2;

    // A 16x4 f32 tile: lane(l15<16) holds row=l15, (K=kp, K=kp+1)
    const float* xrow = x  + (rowBase + l15) * F_IN + kp;
    // B 4x16 tile (Wc^T): lane holds col n=nBase+l15, (K=kp, K=kp+1)
    const float* wrow = Wc + (nBase   + l15) * F_IN + kp;

    v8f c = {0.f, 0.f, 0.f, 0.f, 0.f, 0.f, 0.f, 0.f};
#pragma unroll
    for (int kk = 0; kk < 16; ++kk) {
        v2f a = *(const v2f*)(xrow + kk * 4);
        v2f b = *(const v2f*)(wrow + kk * 4);
        c = __builtin_amdgcn_wmma_f32_16x16x4_f32(
                /*neg_a=*/false, a, /*neg_b=*/false, b,
                /*c_mod=*/(short)0, c, /*reuse_a=*/false, /*reuse_b=*/false);
    }

    // C/D layout: VGPR v, lanes 0-15 -> M=v, lanes 16-31 -> M=v+8, N=l15
    float* yo = y + (size_t)(rowBase + 8 * hi) * 128 + nBase + l15;
#pragma unroll
    for (int v = 0; v < 8; ++v)
        yo[(size_t)v * 128] = c[v];
}

// per edge: out[dst] += y2[src] (64 coalesced f32 atomics), cnt[dst] += 1
__global__ __launch_bounds__(256) void edge_scatter(const int* __restrict__ ei,
                                                    const float* __restrict__ y,
                                                    float* __restrict__ out,
                                                    float* __restrict__ cnt) {
    int t = blockIdx.x * 256 + threadIdx.x;
    int e = t >> 6;                 // 4 edges per block
    int f = t & 63;
    if (e >= N_EDGES) return;
    int src = ei[e];                // edge_index[0][e]
    int dst = ei[N_EDGES + e];      // edge_index[1][e]
    float val = y[(size_t)src * 128 + 64 + f];   // y2[src][f]
    atomicAdd(&out[(size_t)dst * F_OUT + f], val);
    if (f == 0) atomicAdd(&cnt[dst], 1.0f);
}

// out[i][f] = cnt>0 ? y1[i][f] + b[f] + acc/cnt : 0
__global__ void finalize(const float* __restrict__ y, const float* __restrict__ b,
                         const float* __restrict__ cnt, float* __restrict__ out) {
    int idx = blockIdx.x * blockDim.x + threadIdx.x;
    if (idx >= N_NODES * F_OUT) return;
    int i = idx >> 6, f = idx & 63;
    float c = cnt[i];
    float s = out[idx];
    out[idx] = (c > 0.0f) ? (y[(size_t)i * 128 + f] + b[f] + s / c) : 0.0f;
}

extern "C" void kernel_launch(void* const* d_in, const int* in_sizes, int n_in,
                              void* d_out, int out_size, void* d_ws, size_t ws_size,
                              hipStream_t stream) {
    const float* x  = (const float*)d_in[0];
    const int*   ei = (const int*)  d_in[1];
    const float* W  = (const float*)d_in[2];
    const float* b  = (const float*)d_in[3];
    float*       out = (float*)d_out;

    float* ws  = (float*)d_ws;
    float* Wc  = ws;                          // 8192 floats
    float* y   = ws + 8192;                   // 50000*128 floats
    float* cnt = y + (size_t)N_NODES * 128;   // 50000 floats

    // 1) combined weight
    prep_weights<<<(128 * 64 + 255) / 256, 256, 0, stream>>>(W, Wc);
    // 2) zero accumulators (every call: replays must not accumulate)
    zero_bufs<<<(N_NODES * F_OUT + 255) / 256, 256, 0, stream>>>(out, cnt);
    // 3) node-level WMMA GEMM: y = x @ Wc^T
    gemm_xWc<<<N_NODES / 16, 256, 0, stream>>>(x, Wc, y);
    // 4) edge scatter: out[dst] += y2[src], cnt[dst] += 1
    edge_scatter<<<(N_EDGES * 64) / 256, 256, 0, stream>>>(ei, y, out, cnt);
    // 5) finalize: mean + y1 + bias, zero for isolated nodes
    finalize<<<(N_NODES * F_OUT + 255) / 256, 256, 0, stream>>>(y, b, cnt, out);
}